// DecompGridv4_78099685310706
// MI455X (gfx1250) — compile-verified
//
#include <hip/hip_runtime.h>
#include <hip/hip_bf16.h>
#include <stdint.h>

typedef float f32x4 __attribute__((ext_vector_type(4)));

typedef __attribute__((address_space(3))) int lds_int_t;
typedef __attribute__((address_space(1))) int glob_int_t;

#define PLANE_R   512
#define PLANE_HW  (512 * 512)
#define PLANE_C   16
#define PTS_PER_BLOCK 64   // 256 threads / 4 lanes-per-point

#if __has_builtin(__builtin_amdgcn_global_load_async_to_lds_b32)
#define HAS_ASYNC_LDS 1
#else
#define HAS_ASYNC_LDS 0
#endif

__device__ __forceinline__ void async_wait0() {
#if __has_builtin(__builtin_amdgcn_s_wait_asynccnt)
  __builtin_amdgcn_s_wait_asynccnt(0);
#else
  asm volatile("s_wait_asynccnt 0x0" ::: "memory");
#endif
}

// align_corners=True bilinear index/weight setup (mirrors the reference:
// floor, frac, clamp i0, then i1 = clamp(i0+1)).
__device__ __forceinline__ void bilin_setup(float g, int& i0, int& i1, float& w) {
  float p = (g + 1.0f) * 0.5f * (float)(PLANE_R - 1);
  float f = floorf(p);
  w = p - f;
  int i = (int)f;
  i0 = min(max(i, 0), PLANE_R - 1);
  i1 = min(i0 + 1, PLANE_R - 1);
}

// Stage this block's point coordinates (3 floats per point) into LDS.
// Uses the CDNA5 async global->LDS path when the toolchain exposes it.
__device__ __forceinline__ void stage_coords(const float* __restrict__ x,
                                             float* sc, int base, int B) {
  const int t = threadIdx.x;
  const int ncoord = min(PTS_PER_BLOCK, B - base) * 3;
  if (t < ncoord) {
    const int gi = base * 3 + t;
#if HAS_ASYNC_LDS
    __builtin_amdgcn_global_load_async_to_lds_b32(
        (glob_int_t*)(x + gi),
        (lds_int_t*)(sc + t), 0, 0);
#else
    sc[t] = x[gi];
#endif
  }
#if HAS_ASYNC_LDS
  async_wait0();
#endif
  __syncthreads();
}

// ---------------------------------------------------------------------------
// Packed-layout gather: 4 lanes per point, each lane owns 4 channels.
// 12 independent b128 gathers per lane, all L2-resident (planes = 48MB < 192MB L2).
// Output written with NT stores so the 128MB stream doesn't evict the planes.
// (Defined first so the disasm snippet shows the async staging path.)
// ---------------------------------------------------------------------------
__global__ void __launch_bounds__(256)
gather_packed(const float* __restrict__ x, const float* __restrict__ planesT,
              float* __restrict__ out, int B) {
  __shared__ float sc[PTS_PER_BLOCK * 3];
  const int base = blockIdx.x * PTS_PER_BLOCK;
  stage_coords(x, sc, base, B);

  const int t = threadIdx.x;
  const int q = t >> 2;        // point within block
  const int sub = t & 3;       // channel quad
  const int pt = base + q;
  if (pt >= B) return;

  const float cx = sc[q * 3 + 0];
  const float cy = sc[q * 3 + 1];
  const float cz = sc[q * 3 + 2];

  // PLANE_DIMIDS = [(0,1),(0,2),(1,2)]; gx indexes width, gy indexes height.
  const float gxs[3] = {cx, cx, cy};
  const float gys[3] = {cy, cz, cz};

  f32x4 v[3][4];
  float w[3][4];
#pragma unroll
  for (int p = 0; p < 3; ++p) {
    int x0, x1, y0, y1;
    float wx, wy;
    bilin_setup(gxs[p], x0, x1, wx);
    bilin_setup(gys[p], y0, y1, wy);
    const float* bp = planesT + (size_t)p * PLANE_HW * PLANE_C + (sub << 2);
    v[p][0] = *(const f32x4*)(bp + ((size_t)((y0 << 9) + x0) << 4));
    v[p][1] = *(const f32x4*)(bp + ((size_t)((y0 << 9) + x1) << 4));
    v[p][2] = *(const f32x4*)(bp + ((size_t)((y1 << 9) + x0) << 4));
    v[p][3] = *(const f32x4*)(bp + ((size_t)((y1 << 9) + x1) << 4));
    const float iwx = 1.0f - wx, iwy = 1.0f - wy;
    w[p][0] = iwx * iwy;
    w[p][1] = wx * iwy;
    w[p][2] = iwx * wy;
    w[p][3] = wx * wy;
  }

  f32x4 f0 = v[0][0] * w[0][0] + v[0][1] * w[0][1] + v[0][2] * w[0][2] + v[0][3] * w[0][3];
  f32x4 f1 = v[1][0] * w[1][0] + v[1][1] * w[1][1] + v[1][2] * w[1][2] + v[1][3] * w[1][3];
  f32x4 f2 = v[2][0] * w[2][0] + v[2][1] * w[2][1] + v[2][2] * w[2][2] + v[2][3] * w[2][3];
  f32x4 acc = f0 * f1 * f2;

  __builtin_nontemporal_store(acc, (f32x4*)(out + (size_t)pt * PLANE_C + (sub << 2)));
}

// ---------------------------------------------------------------------------
// Repack (C,H,W) -> (H,W,C) so a texel's 16 channels are one 64B cache line.
// Reads are lane-coalesced per channel (non-temporal: read-once).
// Writes are 64B contiguous per thread and stay L2-resident for the gather.
// ---------------------------------------------------------------------------
__global__ void __launch_bounds__(256)
repack_kernel(const float* __restrict__ p0, const float* __restrict__ p1,
              const float* __restrict__ p2, float* __restrict__ dst) {
  const int plane = blockIdx.y;
  const float* src = (plane == 0) ? p0 : (plane == 1) ? p1 : p2;
  float* d = dst + (size_t)plane * PLANE_HW * PLANE_C;

  const int tex = blockIdx.x * blockDim.x + threadIdx.x;  // 0 .. HW-1
  float v[PLANE_C];
#pragma unroll
  for (int c = 0; c < PLANE_C; ++c)
    v[c] = __builtin_nontemporal_load(src + (size_t)c * PLANE_HW + tex);

  f32x4* o = (f32x4*)(d + (size_t)tex * PLANE_C);
#pragma unroll
  for (int k = 0; k < 4; ++k) {
    f32x4 t;
    t[0] = v[4 * k + 0];
    t[1] = v[4 * k + 1];
    t[2] = v[4 * k + 2];
    t[3] = v[4 * k + 3];
    o[k] = t;
  }
}

// ---------------------------------------------------------------------------
// Fallback: gather straight from the original (C,H,W) layout (used only if
// the workspace is too small for the repacked planes).
// ---------------------------------------------------------------------------
__device__ __forceinline__ f32x4 sample4_direct(const float* __restrict__ P,
                                                float gx, float gy, int sub) {
  int x0, x1, y0, y1;
  float wx, wy;
  bilin_setup(gx, x0, x1, wx);
  bilin_setup(gy, y0, y1, wy);
  const int o00 = (y0 << 9) + x0, o01 = (y0 << 9) + x1;
  const int o10 = (y1 << 9) + x0, o11 = (y1 << 9) + x1;
  const float iwx = 1.0f - wx, iwy = 1.0f - wy;
  const float w00 = iwx * iwy, w01 = wx * iwy, w10 = iwx * wy, w11 = wx * wy;
  f32x4 r;
#pragma unroll
  for (int k = 0; k < 4; ++k) {
    const float* ch = P + (size_t)(sub * 4 + k) * PLANE_HW;
    r[k] = ch[o00] * w00 + ch[o01] * w01 + ch[o10] * w10 + ch[o11] * w11;
  }
  return r;
}

__global__ void __launch_bounds__(256)
gather_direct(const float* __restrict__ x, const float* __restrict__ p0,
              const float* __restrict__ p1, const float* __restrict__ p2,
              float* __restrict__ out, int B) {
  __shared__ float sc[PTS_PER_BLOCK * 3];
  const int base = blockIdx.x * PTS_PER_BLOCK;
  stage_coords(x, sc, base, B);

  const int t = threadIdx.x;
  const int q = t >> 2;
  const int sub = t & 3;
  const int pt = base + q;
  if (pt >= B) return;

  const float cx = sc[q * 3 + 0];
  const float cy = sc[q * 3 + 1];
  const float cz = sc[q * 3 + 2];

  f32x4 acc = sample4_direct(p0, cx, cy, sub);
  acc *= sample4_direct(p1, cx, cz, sub);
  acc *= sample4_direct(p2, cy, cz, sub);

  __builtin_nontemporal_store(acc, (f32x4*)(out + (size_t)pt * PLANE_C + (sub << 2)));
}

extern "C" void kernel_launch(void* const* d_in, const int* in_sizes, int n_in,
                              void* d_out, int out_size, void* d_ws, size_t ws_size,
                              hipStream_t stream) {
  const float* x  = (const float*)d_in[0];
  const float* p0 = (const float*)d_in[1];
  const float* p1 = (const float*)d_in[2];
  const float* p2 = (const float*)d_in[3];
  float* out = (float*)d_out;

  const int B = in_sizes[0] / 3;
  const int nblk = (B + PTS_PER_BLOCK - 1) / PTS_PER_BLOCK;
  const size_t need = (size_t)3 * PLANE_HW * PLANE_C * sizeof(float);  // 48 MB

  if (ws_size >= need) {
    dim3 rg(PLANE_HW / 256, 3);
    repack_kernel<<<rg, 256, 0, stream>>>(p0, p1, p2, (float*)d_ws);
    gather_packed<<<nblk, 256, 0, stream>>>(x, (const float*)d_ws, out, B);
  } else {
    gather_direct<<<nblk, 256, 0, stream>>>(x, p0, p1, p2, out, B);
  }
}